// Attention_14955076125505
// MI455X (gfx1250) — compile-verified
//
#include <hip/hip_runtime.h>
#include <hip/hip_bf16.h>

// ---------------------------------------------------------------------------
// Problem constants (match reference)
// ---------------------------------------------------------------------------
constexpr int NB    = 8;     // batch
constexpr int NC    = 512;   // input channels
constexpr int NN    = 1024;  // H*W tokens
constexpr int NOUT  = 512;   // oup
constexpr int HEADS = 16;
constexpr int DH    = 32;    // dim_head
constexpr float SCALE = 0.17677669529663687f;  // 32^-0.5

typedef __attribute__((ext_vector_type(2))) float v2f;
typedef __attribute__((ext_vector_type(8))) float v8f;
typedef __attribute__((ext_vector_type(4))) int   v4i;

// D = A(16x4 f32) * B(4x16 f32) + C(16x16 f32)
__device__ __forceinline__ v8f wmma_f32(v2f a, v2f b, v8f c) {
    return __builtin_amdgcn_wmma_f32_16x16x4_f32(
        /*neg_a=*/false, a, /*neg_b=*/false, b,
        /*c_mod=*/(short)0, c, /*reuse_a=*/false, /*reuse_b=*/false);
}

// ---------------------------------------------------------------------------
// 16-byte global -> LDS copy. Prefer the gfx1250 async-to-LDS path
// (GLOBAL_LOAD_ASYNC_TO_LDS_B128, tracked by ASYNCcnt, no VGPR round-trip);
// fall back to load+ds_store if the builtin is unavailable.
// Builtin signature (from hipcc diagnostic): (v4i AS1*, v4i AS3*, imm, imm).
// ---------------------------------------------------------------------------
#ifndef __has_builtin
#define __has_builtin(x) 0
#endif
#if __has_builtin(__builtin_amdgcn_global_load_async_to_lds_b128)
#define ASYNC_LDS_COPY 1
#else
#define ASYNC_LDS_COPY 0
#endif

__device__ __forceinline__ void copy16_g2lds(const float* gsrc, float* ldst) {
#if ASYNC_LDS_COPY
    __builtin_amdgcn_global_load_async_to_lds_b128(
        (__attribute__((address_space(1))) v4i*)gsrc,
        (__attribute__((address_space(3))) v4i*)ldst,
        /*offset=*/0, /*cpol=*/0);
#else
    const float4 v = *(const float4*)gsrc;
    *(float4*)ldst = v;
#endif
}

__device__ __forceinline__ void copy_g2lds_wait() {
#if ASYNC_LDS_COPY
#if __has_builtin(__builtin_amdgcn_s_wait_asynccnt)
    __builtin_amdgcn_s_wait_asynccnt(0);
#else
    asm volatile("s_wait_asynccnt 0x0" ::: "memory");
#endif
#endif
}

// ---------------------------------------------------------------------------
// Kernel 1: QKV projection. One wave computes a 16(n) x 64(o) stripe, K=512.
//   qkv[b,n,o] = sum_c x[b,c,n] * w_qkv[o,c]     (o in [0,1536))
// The 16(n) x 512(c) x-stripe is staged once in LDS (32KB) via async copies;
// 4 o-tiles share each A fragment. Results scatter into head-major Q/K/V.
// ---------------------------------------------------------------------------
__global__ __launch_bounds__(32)
void qkv_proj_kernel(const float* __restrict__ x, const float* __restrict__ wqkv,
                     float* __restrict__ Q, float* __restrict__ Kv,
                     float* __restrict__ Vv) {
    __shared__ float Sx[NC * 16];       // [c][n16]  32768 bytes

    const int lane = threadIdx.x;
    const int m  = lane & 15;           // A row / B col within tile
    const int hi = lane >> 4;           // half-wave selects K pair
    const int ks = hi * 2;
    const int o0 = blockIdx.x * 64;     // 0..1535 step 64
    const int n0 = blockIdx.y * 16;     // 0..1023 step 16
    const int b  = blockIdx.z;

    const float* xb = x + (size_t)b * NC * NN;    // x[b][c][n], n contiguous

    // Stage x[b, 0:512, n0:n0+16] -> LDS, 16B granules, fully coalesced.
    for (int it = 0; it < (NC * 16) / (32 * 4); ++it) {   // 64 iterations
        const int idx = it * 32 + lane;                   // 0..2047
        const int c = idx >> 2;
        const int q = idx & 3;
        copy16_g2lds(xb + (size_t)c * NN + n0 + q * 4, &Sx[c * 16 + q * 4]);
    }
    copy_g2lds_wait();
    __syncthreads();

    v8f acc[4] = {};
    for (int k0 = 0; k0 < NC; k0 += 4) {
        v2f a;                                            // A[m,k] from LDS
        a.x = Sx[(k0 + ks) * 16 + m];
        a.y = Sx[(k0 + ks + 1) * 16 + m];
#pragma unroll
        for (int t = 0; t < 4; ++t) {                     // B[k,n] contiguous pair
            v2f bf;
            const float* wp = wqkv + (size_t)(o0 + 16 * t + m) * NC + k0 + ks;
            bf.x = wp[0];
            bf.y = wp[1];
            acc[t] = wmma_f32(a, bf, acc[t]);
        }
    }
    // C/D layout: VGPR r -> row (r + 8*hi), col = lane&15
#pragma unroll
    for (int t = 0; t < 4; ++t) {
        const int ob = o0 + 16 * t;
        const int which = ob >> 9;                        // 0:Q 1:K 2:V (uniform)
        float* dst = (which == 0) ? Q : (which == 1) ? Kv : Vv;
        const int oo = (ob & 511) + m;
        const int h  = oo >> 5;
        const int d  = oo & 31;
#pragma unroll
        for (int r = 0; r < 8; ++r) {
            const int n = n0 + r + 8 * hi;
            dst[((((size_t)b * HEADS + h) * NN) + n) * DH + d] = acc[t][r];
        }
    }
}

// ---------------------------------------------------------------------------
// Kernel 2: attention for one (b, head, 16-row q block). Single wave.
// Full 16x1024 score stripe staged in LDS (64KB), exact softmax, then P @ V.
// ---------------------------------------------------------------------------
__global__ __launch_bounds__(32)
void attn_kernel(const float* __restrict__ Q, const float* __restrict__ Kv,
                 const float* __restrict__ Vv,
                 const float* __restrict__ bias_table,
                 const int* __restrict__ rel_index,
                 float* __restrict__ AO) {
    __shared__ float S[16 * NN];            // 65536 bytes

    const int lane = threadIdx.x;
    const int m  = lane & 15;
    const int hi = lane >> 4;
    const int ks = hi * 2;
    const int n0 = blockIdx.x * 16;
    const int h  = blockIdx.y;
    const int b  = blockIdx.z;

    const size_t bh = ((size_t)b * HEADS + h) * NN;
    const float* Qb = Q  + bh * DH;
    const float* Kb = Kv + bh * DH;
    const float* Vb = Vv + bh * DH;

    // Q tile as A fragments for all 8 K-steps (d = 0..31)
    v2f qa[8];
#pragma unroll
    for (int j = 0; j < 8; ++j) {
        const float* qp = Qb + (size_t)(n0 + m) * DH + 4 * j + ks;
        qa[j].x = qp[0];
        qa[j].y = qp[1];
    }

    // ---- Phase 1: scores + relative bias -> LDS -------------------------
    for (int mt = 0; mt < NN / 16; ++mt) {
        const int m0 = mt * 16;
        v8f acc = {};
#pragma unroll
        for (int j = 0; j < 8; ++j) {
            v2f kb;  // B[k=d, n=mcol] = K[m0+n, d]
            const float* kp = Kb + (size_t)(m0 + m) * DH + 4 * j + ks;
            kb.x = kp[0];
            kb.y = kp[1];
            acc = wmma_f32(qa[j], kb, acc);
        }
#pragma unroll
        for (int r = 0; r < 8; ++r) {
            const int row = r + 8 * hi;          // local q row 0..15
            const int col = m0 + m;
            const int idx = rel_index[(size_t)(n0 + row) * NN + col];
            const float bias = bias_table[(size_t)idx * HEADS + h];
            S[row * NN + col] = acc[r] * SCALE + bias;
        }
    }
    __syncthreads();

    // ---- Phase 2: exact softmax stats. lane -> row (lane&15), half hi ---
    float* Srow = S + m * NN + hi * (NN / 2);
    float rmax = -3.4e38f;
    for (int i = 0; i < NN / 2; i += 4) {
        const float4 v4 = *(const float4*)(Srow + i);
        rmax = fmaxf(rmax, fmaxf(fmaxf(v4.x, v4.y), fmaxf(v4.z, v4.w)));
    }
    rmax = fmaxf(rmax, __shfl_xor(rmax, 16, 32));
    float rsum = 0.f;
    for (int i = 0; i < NN / 2; i += 4) {
        float4 v4 = *(float4*)(Srow + i);
        v4.x = __expf(v4.x - rmax);
        v4.y = __expf(v4.y - rmax);
        v4.z = __expf(v4.z - rmax);
        v4.w = __expf(v4.w - rmax);
        *(float4*)(Srow + i) = v4;               // keep un-normalized P in LDS
        rsum += v4.x + v4.y + v4.z + v4.w;
    }
    rsum += __shfl_xor(rsum, 16, 32);
    const float rinv = 1.0f / rsum;              // lane l holds row (l&15)
    __syncthreads();

    // ---- Phase 3: O = P @ V  (two 16x16 d-tiles, normalize at the end) --
    v8f oa = {}, ob = {};
    for (int mt = 0; mt < NN / 16; ++mt) {
        const int m0 = mt * 16;
        v2f pa[4];
#pragma unroll
        for (int j = 0; j < 4; ++j) {            // A[m=row, k=mcol] from LDS
            const float* pp = S + m * NN + m0 + 4 * j + ks;
            pa[j].x = pp[0];
            pa[j].y = pp[1];
        }
#pragma unroll
        for (int j = 0; j < 4; ++j) {
            v2f v0, v1;                          // B[k=mcol, n=d]
            const float* vp = Vb + (size_t)(m0 + 4 * j + ks) * DH;
            v0.x = vp[m];        v0.y = vp[DH + m];
            v1.x = vp[16 + m];   v1.y = vp[DH + 16 + m];
            oa = wmma_f32(pa[j], v0, oa);
            ob = wmma_f32(pa[j], v1, ob);
        }
    }
    // AO[b, n, oup]  with oup = h*32 + d
    const int oc = h * DH;
#pragma unroll
    for (int r = 0; r < 8; ++r) {
        const int row = r + 8 * hi;
        const float sc = __shfl(rinv, row, 32);
        const size_t base = ((size_t)b * NN + (n0 + row)) * NOUT + oc;
        AO[base + m]      = oa[r] * sc;
        AO[base + 16 + m] = ob[r] * sc;
    }
}

// ---------------------------------------------------------------------------
// Kernel 3: output projection + bias, transposed store out[b, o, n].
//   out[b,o,n] = sum_c AO[b,n,c] * w_out[o,c] + b_out[o]
// One wave computes a 16(n) x 64(o) stripe; A fragments (contiguous b64 loads)
// are shared by the 4 o-tiles.
// ---------------------------------------------------------------------------
__global__ __launch_bounds__(32)
void out_proj_kernel(const float* __restrict__ AO, const float* __restrict__ wout,
                     const float* __restrict__ bout, float* __restrict__ out) {
    const int lane = threadIdx.x;
    const int m  = lane & 15;
    const int hi = lane >> 4;
    const int ks = hi * 2;
    const int o0 = blockIdx.x * 64;
    const int n0 = blockIdx.y * 16;
    const int b  = blockIdx.z;

    const float* Ab = AO + (size_t)b * NN * NOUT;
    v8f acc[4] = {};
    for (int k0 = 0; k0 < NOUT; k0 += 4) {
        v2f a;
        const float* ap = Ab + (size_t)(n0 + m) * NOUT + k0 + ks;   // contiguous
        a.x = ap[0];
        a.y = ap[1];
#pragma unroll
        for (int t = 0; t < 4; ++t) {
            v2f bf;
            const float* wp = wout + (size_t)(o0 + 16 * t + m) * NOUT + k0 + ks;
            bf.x = wp[0];
            bf.y = wp[1];
            acc[t] = wmma_f32(a, bf, acc[t]);
        }
    }
#pragma unroll
    for (int t = 0; t < 4; ++t) {
        const int o = o0 + 16 * t + m;
        const float bias = bout[o];
#pragma unroll
        for (int r = 0; r < 8; ++r) {
            const int n = n0 + r + 8 * hi;
            out[((size_t)b * NOUT + o) * NN + n] = acc[t][r] + bias;
        }
    }
}

// ---------------------------------------------------------------------------
// Host-side launcher
// ---------------------------------------------------------------------------
extern "C" void kernel_launch(void* const* d_in, const int* in_sizes, int n_in,
                              void* d_out, int out_size, void* d_ws, size_t ws_size,
                              hipStream_t stream) {
    const float* x          = (const float*)d_in[0];
    const float* w_qkv      = (const float*)d_in[1];
    const float* w_out      = (const float*)d_in[2];
    const float* b_out      = (const float*)d_in[3];
    const float* bias_table = (const float*)d_in[4];
    const int*   rel_index  = (const int*)d_in[5];
    float* out = (float*)d_out;

    const size_t QSZ = (size_t)NB * HEADS * NN * DH;   // 4,194,304 floats
    float* Q  = (float*)d_ws;
    float* K  = Q + QSZ;
    float* V  = K + QSZ;
    float* AO = V + QSZ;                               // NB*NN*NOUT floats

    // 1) QKV projection: 24 o-stripes x 64 n-tiles x 8 batches
    qkv_proj_kernel<<<dim3(3 * NOUT / 64, NN / 16, NB), dim3(32), 0, stream>>>(
        x, w_qkv, Q, K, V);

    // 2) Attention: 64 q-blocks x 16 heads x 8 batches
    attn_kernel<<<dim3(NN / 16, HEADS, NB), dim3(32), 0, stream>>>(
        Q, K, V, bias_table, rel_index, AO);

    // 3) Output projection: 8 o-stripes x 64 n-tiles x 8 batches
    out_proj_kernel<<<dim3(NOUT / 64, NN / 16, NB), dim3(32), 0, stream>>>(
        AO, w_out, b_out, out);
}